// GCN_57286273794678
// MI455X (gfx1250) — compile-verified
//
#include <hip/hip_runtime.h>

// ---------------------------------------------------------------------------
// GCN forward for MI455X (gfx1250, wave32).
//   layer: h = relu( Agg( X @ W ) + b ),  Agg uses symmetric norm + self loops
//   head : out = X @ Wc + bc
// GEMMs use V_WMMA_F32_16X16X4_F32 (exact fp32 matrix path on CDNA5).
// B-fragments are pre-packed in LDS as float2 (consecutive-K pairs) so each
// fragment is one aligned ds_load_b64 into an even VGPR pair (no mov fixups).
// Aggregation uses native fp32 global atomics; activations are L2-resident
// (25.6MB << 192MB L2) so the random gather/scatter stays on-chip.
// ---------------------------------------------------------------------------

typedef float v2f __attribute__((ext_vector_type(2)));
typedef float v8f __attribute__((ext_vector_type(8)));

#define NHID 64
#define NFEAT 128
#define NCLS 40

// ---------------- elementwise / degree kernels ----------------

__global__ void k_fill1(float* __restrict__ p, int n) {
    int i = blockIdx.x * blockDim.x + threadIdx.x;
    if (i < n) p[i] = 1.0f;  // self-loop contributes 1 to every degree
}

__global__ void k_deg_count(const int* __restrict__ cols, float* __restrict__ deg, int ne) {
    int i = blockIdx.x * blockDim.x + threadIdx.x;
    if (i < ne) unsafeAtomicAdd(&deg[cols[i]], 1.0f);
}

__global__ void k_rsqrt_inplace(float* __restrict__ d, int n) {
    int i = blockIdx.x * blockDim.x + threadIdx.x;
    if (i < n) d[i] = __frsqrt_rn(d[i]);  // deg >= 1 always (self loops)
}

// out[i,f] = h[i,f] * dinv[i]^2   (the self-loop term initializes the sum)
__global__ void k_agg_init(const float* __restrict__ h, const float* __restrict__ dinv,
                           float* __restrict__ out, long total) {
    long idx = (long)blockIdx.x * blockDim.x + threadIdx.x;
    if (idx >= total) return;
    int i = (int)(idx >> 6);
    float di = dinv[i];
    out[idx] = h[idx] * di * di;
}

// out[col,f] += h[row,f] * dinv[row]*dinv[col]; 64 threads (2 waves) per edge,
// fully coalesced over f; atomics land in L2 (activations are L2-resident).
__global__ void k_agg_edges(const float* __restrict__ h,
                            const int* __restrict__ rows,
                            const int* __restrict__ cols,
                            const float* __restrict__ dinv,
                            float* __restrict__ out, long ne) {
    long idx = (long)blockIdx.x * blockDim.x + threadIdx.x;
    long e = idx >> 6;
    if (e >= ne) return;
    int f = (int)(idx & 63);
    int r = rows[e];
    int c = cols[e];
    float nrm = dinv[r] * dinv[c];
    float v = h[(long)r * NHID + f] * nrm;
    unsafeAtomicAdd(out + (long)c * NHID + f, v);
}

__global__ void k_bias_relu(float* __restrict__ x, const float* __restrict__ b, long total) {
    long idx = (long)blockIdx.x * blockDim.x + threadIdx.x;
    if (idx >= total) return;
    float v = x[idx] + b[idx & 63];
    x[idx] = v > 0.0f ? v : 0.0f;
}

// ---------------- WMMA fp32 GEMM: Y[nrows x fout] = X[nrows x FIN] @ W + bias ----
// One wave owns a 16-row tile and NT column-tiles of 16 (fout <= NT*16).
// A frag (16x4 f32): lanes 0-15 hold row M=lane, K=k..k+1; lanes 16-31 K=k+2..k+3.
// B frag (4x16 f32): VGPR0 lanes{0-15,16-31} = K{k,k+2}; VGPR1 = K{k+1,k+3},
//   i.e. lane needs float2 {W[k+2*hi][n], W[k+2*hi+1][n]} -> pre-packed in LDS
//   as Wp[(k/2 + hi)*FOUTP + n] so it is a single aligned ds_load_b64.
// C/D (16x16 f32): VGPR r, lanes 0-15 -> (M=r, N=lane); lanes 16-31 -> (M=r+8).

template <int FIN, int NT, bool BIAS>
__global__ void k_gemm_wmma(const float* __restrict__ X, const float* __restrict__ W,
                            const float* __restrict__ bias, float* __restrict__ Y,
                            int nrows, int fout) {
    constexpr int FOUTP = NT * 16;
    __shared__ v2f Wp[(FIN / 2) * FOUTP];  // packed consecutive-K pairs
    for (int i = threadIdx.x; i < (FIN / 2) * FOUTP; i += blockDim.x) {
        int kp = i / FOUTP, c = i - kp * FOUTP;
        v2f w;
        w.x = (c < fout) ? W[(2 * kp + 0) * fout + c] : 0.0f;  // zero-pad columns
        w.y = (c < fout) ? W[(2 * kp + 1) * fout + c] : 0.0f;
        Wp[i] = w;
    }
    __syncthreads();

    const int wave = threadIdx.x >> 5;
    const int lane = threadIdx.x & 31;
    const int lo = lane & 15;
    const int hi = lane >> 4;

    const long tile = (long)blockIdx.x * (blockDim.x >> 5) + wave;  // 16-row tile
    if (tile * 16 >= nrows) return;  // wave-uniform: EXEC stays all-1s for WMMA

    const float* Xr = X + tile * 16 * FIN + (long)lo * FIN + 2 * hi;
    const v2f* Wb = Wp + hi * FOUTP + lo;  // + t*16 per tile, + (k/2)*FOUTP per step

    v8f acc[NT];
    const v8f vzero = {0.f, 0.f, 0.f, 0.f, 0.f, 0.f, 0.f, 0.f};
#pragma unroll
    for (int t = 0; t < NT; ++t) acc[t] = vzero;

#pragma unroll 4
    for (int k = 0; k < FIN; k += 4) {
        v2f a = *(const v2f*)(Xr + k);
        const v2f* Wk = Wb + (k >> 1) * FOUTP;
#pragma unroll
        for (int t = 0; t < NT; ++t) {
            v2f b = Wk[t * 16];
            acc[t] = __builtin_amdgcn_wmma_f32_16x16x4_f32(
                false, a, false, b, (short)0, acc[t], false, false);
        }
    }

    float* Yr = Y + tile * 16 * (long)fout;
#pragma unroll
    for (int t = 0; t < NT; ++t) {
        int n = t * 16 + lo;
        if (n < fout) {
            float bv = BIAS ? bias[n] : 0.0f;
#pragma unroll
            for (int r = 0; r < 8; ++r) {
                Yr[(long)(r + 8 * hi) * fout + n] = acc[t][r] + bv;
            }
        }
    }
}

// ---------------------------------------------------------------------------

static inline int ceil_div_l(long a, long b) { return (int)((a + b - 1) / b); }

extern "C" void kernel_launch(void* const* d_in, const int* in_sizes, int n_in,
                              void* d_out, int out_size, void* d_ws, size_t ws_size,
                              hipStream_t stream) {
    const float* x   = (const float*)d_in[0];
    const int*   ei  = (const int*)d_in[1];   // (2, E) int32
    const float* W1  = (const float*)d_in[2];
    const float* b1  = (const float*)d_in[3];
    const float* W2  = (const float*)d_in[4];
    const float* b2  = (const float*)d_in[5];
    const float* Wc  = (const float*)d_in[6];
    const float* bc  = (const float*)d_in[7];
    float* out = (float*)d_out;

    const int N = in_sizes[0] / NFEAT;   // 100000
    const int E = in_sizes[1] / 2;       // 1600000
    const int* rows = ei;
    const int* cols = ei + E;

    // workspace layout: dinv[N] | bufA[N*64] | bufB[N*64]
    float* dinv = (float*)d_ws;
    float* bufA = dinv + (((size_t)N + 255) & ~(size_t)255);
    float* bufB = bufA + (size_t)N * NHID;

    const long totH = (long)N * NHID;
    const int thr = 256;
    const int tiles = (N + 15) / 16;
    const int gemm_blocks = (tiles + 7) / 8;  // 8 waves (wave32) per 256-thr block

    // degrees -> dinv = rsqrt(deg)   (deg includes self loop)
    k_fill1<<<ceil_div_l(N, thr), thr, 0, stream>>>(dinv, N);
    k_deg_count<<<ceil_div_l(E, thr), thr, 0, stream>>>(cols, dinv, E);
    k_rsqrt_inplace<<<ceil_div_l(N, thr), thr, 0, stream>>>(dinv, N);

    // ---- layer 1: h1 = relu(Agg(x @ W1) + b1)
    k_gemm_wmma<NFEAT, 4, false><<<gemm_blocks, thr, 0, stream>>>(x, W1, nullptr, bufA, N, NHID);
    k_agg_init<<<ceil_div_l(totH, thr), thr, 0, stream>>>(bufA, dinv, bufB, totH);
    k_agg_edges<<<ceil_div_l((long)E * 64, thr), thr, 0, stream>>>(bufA, rows, cols, dinv, bufB, E);
    k_bias_relu<<<ceil_div_l(totH, thr), thr, 0, stream>>>(bufB, b1, totH);

    // ---- layer 2: h2 = relu(Agg(h1 @ W2) + b2)
    k_gemm_wmma<NHID, 4, false><<<gemm_blocks, thr, 0, stream>>>(bufB, W2, nullptr, bufA, N, NHID);
    k_agg_init<<<ceil_div_l(totH, thr), thr, 0, stream>>>(bufA, dinv, bufB, totH);
    k_agg_edges<<<ceil_div_l((long)E * 64, thr), thr, 0, stream>>>(bufA, rows, cols, dinv, bufB, E);
    k_bias_relu<<<ceil_div_l(totH, thr), thr, 0, stream>>>(bufB, b2, totH);

    // ---- head: out = h2 @ Wc + bc   (40 cols padded to 48 = 3 tiles)
    k_gemm_wmma<NHID, 3, true><<<gemm_blocks, thr, 0, stream>>>(bufB, Wc, bc, out, N, NCLS);
}